// UNet_Vit_91293824844222
// MI455X (gfx1250) — compile-verified
//
#include <hip/hip_runtime.h>
#include <math.h>

// ---------------------------------------------------------------------------
// UNet_Vit superpixel head for MI455X (gfx1250, wave32).
//   - scatter-mean is HBM/atomic bound: stream 140MB once, f32 atomics to L2
//   - Conv1d(64->1,k=3) lowered to WMMA f32 16x16x4 GEMM: z = W(1x192) * G(192x2048)
// ---------------------------------------------------------------------------

typedef __attribute__((ext_vector_type(2))) float v2f;
typedef __attribute__((ext_vector_type(8))) float v8f;

#define HW       1048576   // 1024*1024 pixels, B=1
#define SUP      2048
#define CF       32
#define C2       64
#define CIN      3
#define EPS_NORM 1e-12f
#define EPS_BN   1e-5f

// ---- workspace layout (float units) ----
#define WS_MSUM  0                         // [SUP*CF]   segment sum of normalized conv feat
#define WS_XSUM  (WS_MSUM + SUP*CF)        // [SUP*CIN]  segment sum of x
#define WS_CNT   (WS_XSUM + SUP*CIN)       // [SUP]      segment counts
#define PADW     2050                      // s = -1..2048 (zero padded ends)
#define WS_ALLF  (WS_CNT + SUP)            // [C2*PADW]  normalized all_F, padded for conv
#define WS_Z     (WS_ALLF + C2*PADW)       // [SUP]      conv1d output
#define WS_TOTAL (WS_Z + SUP)

// ---- output layout: (result[2048], gap[1], all_F[64*2048], X_F[3*2048]) ----
#define OUT_RESULT 0
#define OUT_GAP    2048
#define OUT_ALLF   2049
#define OUT_XF     (OUT_ALLF + C2*SUP)

// ---------------------------------------------------------------------------
// Kernel A: per-pixel channel L2-norm of conv_feat + scatter-add into segment
// accumulators. 36 global f32 atomics per pixel; accumulators stay hot in L2.
// ---------------------------------------------------------------------------
__global__ void scatter_kernel(const float* __restrict__ x,
                               const float* __restrict__ conv,
                               const int*   __restrict__ seg,
                               float*       __restrict__ ws) {
    int p = blockIdx.x * blockDim.x + threadIdx.x;
    if (p >= HW) return;

    float v[CF];
    float ss = 0.f;
#pragma unroll
    for (int c = 0; c < CF; ++c) {
        float t = conv[c * HW + p];   // coalesced within each channel plane
        v[c] = t;
        ss += t * t;
    }
    float inv = 1.f / fmaxf(sqrtf(ss), EPS_NORM);

    int s = seg[p];
    float* msum = ws + WS_MSUM + (size_t)s * CF;
#pragma unroll
    for (int c = 0; c < CF; ++c) atomicAdd(&msum[c], v[c] * inv);

    float* xsum = ws + WS_XSUM + (size_t)s * CIN;
#pragma unroll
    for (int c = 0; c < CIN; ++c) atomicAdd(&xsum[c], x[c * HW + p]);

    atomicAdd(ws + WS_CNT + s, 1.f);
}

// ---------------------------------------------------------------------------
// Kernel B: one wave32 per segment. Segment means, vit L2-norm (32ch),
// joint 64-channel L2-norm; emit all_F / X_F outputs + padded GEMM operand.
// ---------------------------------------------------------------------------
__device__ inline float wave_sum32(float v) {
#pragma unroll
    for (int m = 16; m > 0; m >>= 1) v += __shfl_xor(v, m, 32);
    return v;
}

__global__ void finalize_kernel(const float* __restrict__ vit,
                                float*       __restrict__ ws,
                                float*       __restrict__ out) {
    int seg  = blockIdx.x * (blockDim.x >> 5) + (threadIdx.x >> 5);
    int lane = threadIdx.x & 31;
    if (seg >= SUP) return;

    float cnt = fmaxf(ws[WS_CNT + seg], 1.f);
    float mf  = ws[WS_MSUM + (size_t)seg * CF + lane] / cnt;

    if (lane < CIN)   // X_F output: [B, Cin, Sup] -> c*SUP + s
        out[OUT_XF + lane * SUP + seg] = ws[WS_XSUM + (size_t)seg * CIN + lane] / cnt;

    float vt  = vit[(size_t)seg * CF + lane];
    float vss = wave_sum32(vt * vt);
    float vn  = vt / fmaxf(sqrtf(vss), EPS_NORM);

    float tot  = wave_sum32(mf * mf + vn * vn);
    float invn = 1.f / fmaxf(sqrtf(tot), EPS_NORM);
    float a0 = mf * invn;    // channel lane
    float a1 = vn * invn;    // channel 32+lane

    out[OUT_ALLF + lane        * SUP + seg] = a0;
    out[OUT_ALLF + (CF + lane) * SUP + seg] = a1;
    ws[WS_ALLF + lane        * PADW + 1 + seg] = a0;   // padded copy for conv GEMM
    ws[WS_ALLF + (CF + lane) * PADW + 1 + seg] = a1;
}

// ---------------------------------------------------------------------------
// Kernel C: Conv1d(64->1,k=3,pad=1) as WMMA f32 16x16x4 GEMM.
//   z[s] = b + sum_{j=0..191} W[j] * G[j][s],  j = k*64 + c,
//   W[j] = w_soft[c*3+k],  G[j][s] = all_F_pad[c][s+k]  (padded index)
// One wave per 16-position tile; A rows are W replicated so every D row == z.
// ISA operand layouts (16x16x4 f32, wave32):
//   A v2f: lanes 0-15 -> K = j0+{0,1}; lanes 16-31 -> K = j0+{2,3}
//   B v2f: same K split, N = lane&15
//   D v8f: VGPR0, lanes 0-15 = row M=0, N = lane
// ---------------------------------------------------------------------------
__global__ void conv1d_wmma_kernel(const float* __restrict__ w,      // [64*3]
                                   const float* __restrict__ bsoft,  // [1]
                                   const float* __restrict__ allf,   // ws + WS_ALLF
                                   float*       __restrict__ z) {
    int wave = blockIdx.x * (blockDim.x >> 5) + (threadIdx.x >> 5);
    int lane = threadIdx.x & 31;
    int half = lane >> 4;
    int ln   = lane & 15;
    if (wave >= SUP / 16) return;        // exact grid: never taken, EXEC stays full
    int s0 = wave * 16;

    v8f acc = {0.f, 0.f, 0.f, 0.f, 0.f, 0.f, 0.f, 0.f};

#pragma unroll 4
    for (int t = 0; t < 48; ++t) {       // K = 192 in chunks of 4
        int jx = 4 * t + 2 * half;
        int jy = jx + 1;
        v2f a, b;
        a.x = w[(jx & 63) * 3 + (jx >> 6)];
        a.y = w[(jy & 63) * 3 + (jy >> 6)];
        b.x = allf[(jx & 63) * PADW + s0 + ln + (jx >> 6)];
        b.y = allf[(jy & 63) * PADW + s0 + ln + (jy >> 6)];
        acc = __builtin_amdgcn_wmma_f32_16x16x4_f32(
                  /*neg_a=*/false, a, /*neg_b=*/false, b,
                  /*c_mod=*/(short)0, acc, /*reuse_a=*/false, /*reuse_b=*/false);
    }

    if (half == 0)                       // row M=0: VGPR0, lanes 0-15
        z[s0 + ln] = acc[0] + bsoft[0];
}

// ---------------------------------------------------------------------------
// Kernel D: BatchNorm1d (training batch stats over 2048) + sigmoid + GAP.
// ---------------------------------------------------------------------------
__global__ void bn_sig_kernel(const float* __restrict__ z,
                              const float* __restrict__ gamma,
                              const float* __restrict__ beta,
                              float*       __restrict__ out) {
    __shared__ float red[1024];
    __shared__ float stats[2];
    int tid = threadIdx.x;
    float z0 = z[tid], z1 = z[tid + 1024];

    red[tid] = z0 + z1;
    __syncthreads();
    for (int o = 512; o > 0; o >>= 1) {
        if (tid < o) red[tid] += red[tid + o];
        __syncthreads();
    }
    if (tid == 0) stats[0] = red[0] * (1.f / 2048.f);
    __syncthreads();

    float mu = stats[0];
    float d0 = z0 - mu, d1 = z1 - mu;
    red[tid] = d0 * d0 + d1 * d1;
    __syncthreads();
    for (int o = 512; o > 0; o >>= 1) {
        if (tid < o) red[tid] += red[tid + o];
        __syncthreads();
    }
    if (tid == 0) stats[1] = red[0] * (1.f / 2048.f);
    __syncthreads();

    float sc = gamma[0] / sqrtf(stats[1] + EPS_BN);
    float r0 = 1.f / (1.f + expf(-(sc * d0 + beta[0])));
    float r1 = 1.f / (1.f + expf(-(sc * d1 + beta[0])));
    out[OUT_RESULT + tid]        = r0;
    out[OUT_RESULT + tid + 1024] = r1;

    red[tid] = r0 + r1;
    __syncthreads();
    for (int o = 512; o > 0; o >>= 1) {
        if (tid < o) red[tid] += red[tid + o];
        __syncthreads();
    }
    if (tid == 0) out[OUT_GAP] = red[0] * (1.f / 2048.f);
}

// ---------------------------------------------------------------------------
extern "C" void kernel_launch(void* const* d_in, const int* in_sizes, int n_in,
                              void* d_out, int out_size, void* d_ws, size_t ws_size,
                              hipStream_t stream) {
    (void)in_sizes; (void)n_in; (void)out_size;
    const float* x     = (const float*)d_in[0];
    const float* conv  = (const float*)d_in[1];
    const float* vit   = (const float*)d_in[2];
    const int*   seg   = (const int*)  d_in[3];
    const float* w     = (const float*)d_in[4];
    const float* bsoft = (const float*)d_in[5];
    const float* gamma = (const float*)d_in[6];
    const float* beta  = (const float*)d_in[7];
    float* out = (float*)d_out;
    float* ws  = (float*)d_ws;

    size_t need = (size_t)WS_TOTAL * sizeof(float);
    if (need > ws_size) need = ws_size;
    hipMemsetAsync(d_ws, 0, need, stream);   // zeroes accumulators + conv padding

    scatter_kernel  <<<HW / 256, 256, 0, stream>>>(x, conv, seg, ws);
    finalize_kernel <<<SUP / 8, 256, 0, stream>>>(vit, ws, out);
    conv1d_wmma_kernel<<<(SUP / 16) / 8, 256, 0, stream>>>(w, bsoft,
                                                           ws + WS_ALLF, ws + WS_Z);
    bn_sig_kernel   <<<1, 1024, 0, stream>>>(ws + WS_Z, gamma, beta, out);
}